// CRPSLoss_51625506898387
// MI455X (gfx1250) — compile-verified
//
#include <hip/hip_runtime.h>
#include <hip/hip_bf16.h>

// CRPS ensemble loss for MI455X (gfx1250, wave32).
// Memory-bound: ~44 MB read @ 23.3 TB/s -> ~2 us floor. Single pass, all 20
// ensemble members held in VGPRs (float4 x 20 = 80 VGPRs/lane), 128-bit loads.
// Final reduction uses v_wmma_f32_16x16x4_f32 with A=ones as an exact
// ones-matmul sum (deterministic, no float atomics).

typedef __attribute__((ext_vector_type(2))) float v2f;
typedef __attribute__((ext_vector_type(8))) float v8f;

#define N_ENS 20
#define MAIN_BLOCK 256
#define MAX_BLOCKS 1024

__device__ __forceinline__ void acc_abs4(float4& s, const float4& a, const float4& b) {
    s.x += fabsf(a.x - b.x);
    s.y += fabsf(a.y - b.y);
    s.z += fabsf(a.z - b.z);
    s.w += fabsf(a.w - b.w);
}

__global__ __launch_bounds__(MAIN_BLOCK) void crps_main_kernel(
    const float* __restrict__ fore,   // (N_ENS, P) float32
    const float* __restrict__ obs,    // (P,) float32
    float* __restrict__ partials,     // (gridDim.x,) float32
    int P) {
    const int   P4     = P >> 2;
    const float inv_n  = 1.0f / (float)N_ENS;
    const float inv_n2 = 1.0f / (float)(N_ENS * N_ENS);
    const int   stride = gridDim.x * blockDim.x;

    float acc = 0.0f;

    const float4* __restrict__ obs4 = reinterpret_cast<const float4*>(obs);

    for (int p = blockIdx.x * blockDim.x + threadIdx.x; p < P4; p += stride) {
        float4 y = obs4[p];

        float4 x[N_ENS];
#pragma unroll
        for (int i = 0; i < N_ENS; ++i) {
            x[i] = reinterpret_cast<const float4*>(fore + (size_t)i * (size_t)P)[p];
        }

        float4 first = make_float4(0.f, 0.f, 0.f, 0.f);
        float4 pair  = make_float4(0.f, 0.f, 0.f, 0.f);
#pragma unroll
        for (int i = 0; i < N_ENS; ++i) {
            acc_abs4(first, x[i], y);
#pragma unroll
            for (int j = i + 1; j < N_ENS; ++j) {
                acc_abs4(pair, x[i], x[j]);
            }
        }
        acc += (first.x + first.y + first.z + first.w) * inv_n
             - (pair.x + pair.y + pair.z + pair.w) * inv_n2;
    }

    // Scalar tail (P not divisible by 4). P=524288 in practice -> no-op.
    const int tail0 = P4 << 2;
    for (int p = tail0 + blockIdx.x * blockDim.x + threadIdx.x; p < P; p += stride) {
        float y = obs[p];
        float xs[N_ENS];
#pragma unroll
        for (int i = 0; i < N_ENS; ++i) xs[i] = fore[(size_t)i * (size_t)P + p];
        float first = 0.f, pairs = 0.f;
#pragma unroll
        for (int i = 0; i < N_ENS; ++i) {
            first += fabsf(xs[i] - y);
#pragma unroll
            for (int j = i + 1; j < N_ENS; ++j) pairs += fabsf(xs[i] - xs[j]);
        }
        acc += first * inv_n - pairs * inv_n2;
    }

    // Block-level tree reduction in LDS.
    __shared__ float sdata[MAIN_BLOCK];
    sdata[threadIdx.x] = acc;
    __syncthreads();
#pragma unroll
    for (int s = MAIN_BLOCK / 2; s > 0; s >>= 1) {
        if ((int)threadIdx.x < s) sdata[threadIdx.x] += sdata[threadIdx.x + s];
        __syncthreads();
    }
    if (threadIdx.x == 0) partials[blockIdx.x] = sdata[0];
}

// Final reduction: 256 threads, one block. Deterministic order.
// The last 32->1 step runs a v_wmma_f32_16x16x4_f32 with A=ones:
// D[m][n] = sum_k B[k][n], so summing row 0 of D over N gives the exact total.
__global__ __launch_bounds__(256) void crps_reduce_kernel(
    const float* __restrict__ partials, int n, float* __restrict__ out, float scale) {
    __shared__ float sdata[256];

    float acc = 0.0f;
    for (int i = threadIdx.x; i < n; i += 256) acc += partials[i];
    sdata[threadIdx.x] = acc;
    __syncthreads();

    if (threadIdx.x < 32) {  // wave-aligned branch: EXEC all-1s in wave 0
        // 256 -> 32: lane t sums sdata[t + 32*k]
        float v = 0.0f;
#pragma unroll
        for (int k = 0; k < 8; ++k) v += sdata[threadIdx.x + 32 * k];

        // Pack the 32 lane values into the B operand (4x16 f32 = 2 VGPRs/lane);
        // the unused half is zero, so column sums over K absorb everything.
        v2f a; a[0] = 1.0f; a[1] = 1.0f;   // A = ones(16x4)
        v2f b; b[0] = v;    b[1] = 0.0f;
        v8f c = {};
        c = __builtin_amdgcn_wmma_f32_16x16x4_f32(
                /*neg_a=*/false, a, /*neg_b=*/false, b,
                /*c_mod=*/(short)0, c, /*reuse_a=*/false, /*reuse_b=*/false);

        // c[0] on lanes 0..15 holds D[0][0..15] = the 16 column sums.
        float r = c[0];
#pragma unroll
        for (int off = 8; off > 0; off >>= 1) r += __shfl_down(r, off, 32);
        if (threadIdx.x == 0) out[0] = r * scale;
    }
}

extern "C" void kernel_launch(void* const* d_in, const int* in_sizes, int n_in,
                              void* d_out, int out_size, void* d_ws, size_t ws_size,
                              hipStream_t stream) {
    const float* fore = (const float*)d_in[0];   // (20, 4, 1, 8, 128, 128) f32
    const float* obs  = (const float*)d_in[1];   // (4, 1, 8, 128, 128) f32
    float*       out  = (float*)d_out;           // scalar f32
    float*  partials  = (float*)d_ws;            // <= MAX_BLOCKS floats

    const int P  = in_sizes[1];                  // 524288 spatial points
    const int P4 = P >> 2;

    int blocks = (P4 + MAIN_BLOCK - 1) / MAIN_BLOCK;
    if (blocks > MAX_BLOCKS) blocks = MAX_BLOCKS;
    if (blocks < 1) blocks = 1;

    crps_main_kernel<<<blocks, MAIN_BLOCK, 0, stream>>>(fore, obs, partials, P);

    const float scale = 1.0f / (float)P;
    crps_reduce_kernel<<<1, 256, 0, stream>>>(partials, blocks, out, scale);
}